// FR_Model_1700807050029
// MI455X (gfx1250) — compile-verified
//
#include <hip/hip_runtime.h>

// Vector types matching gfx1250 WMMA operand register counts.
typedef __attribute__((ext_vector_type(2))) float v2f;   // A/B frag: 16x4 / 4x16 f32 = 2 VGPRs
typedef __attribute__((ext_vector_type(4))) float v4f;   // staging loads
typedef __attribute__((ext_vector_type(8))) float v8f;   // C/D frag: 16x16 f32 = 8 VGPRs

#define B_DIM   1024
#define Y_DIM   256
#define P_LEN   1024
#define PRE_WIN 48
#define KC      64                       // K-chunk staged through LDS
#define WAVES   8
#define BLK_M   (WAVES * 16)             // 128 batch rows per block
#define SLICE   (KC * PRE_WIN)           // 3072 floats = 12 KB per buffer

__global__ __launch_bounds__(256)
void fr_bjp_jpw_bwj_wmma(const float* __restrict__ x,
                         const float* __restrict__ w,
                         const float* __restrict__ bias,
                         float* __restrict__ out) {
    // Double-buffered, swizzled weight slice. Element (row r, col c) of the
    // KCx48 slice lives at (r>>1)*96 + c*2 + (r&1): a lane's (K,K+1) B-fragment
    // pair is contiguous -> single ds_load_b64, no VGPR repacking.
    __shared__ float ws[2 * SLICE];

    const int tid   = threadIdx.x;
    const int lane  = tid & 31;
    const int wv    = tid >> 5;
    const int j     = blockIdx.y;
    const int m0    = blockIdx.x * BLK_M + wv * 16;  // this wave's 16-row M tile
    const int l15   = lane & 15;
    const int khalf = lane >> 4;                     // 0: lanes 0-15, 1: lanes 16-31

    // x[b][j][p]; this lane streams row (m0+l15) of channel j
    const float* xr = x + ((size_t)(m0 + l15) * Y_DIM + j) * P_LEN;
    const float* wj = w + (size_t)j * P_LEN * PRE_WIN;

    // Per-thread swizzled LDS store offsets (loop-invariant): 3 passes of
    // float4; 4 consecutive elements never cross a 48-col row (4 | 48).
    int ldsoff[3];
#pragma unroll
    for (int t = 0; t < 3; ++t) {
        const int i  = t * 1024 + tid * 4;
        const int ri = i / PRE_WIN;
        const int ci = i % PRE_WIN;
        ldsoff[t] = (ri >> 1) * 96 + ci * 2 + (ri & 1);
    }

    // Init accumulators with bias[j][N] (bias is per-N, identical for every M row)
    const float b0 = bias[j * PRE_WIN +  0 + l15];
    const float b1 = bias[j * PRE_WIN + 16 + l15];
    const float b2 = bias[j * PRE_WIN + 32 + l15];
    v8f c0, c1, c2;
#pragma unroll
    for (int i = 0; i < 8; ++i) { c0[i] = b0; c1[i] = b1; c2[i] = b2; }

    // Prologue: stage chunk 0 into buffer 0 (coalesced b128 global loads).
    v4f rg[3];
#pragma unroll
    for (int t = 0; t < 3; ++t)
        rg[t] = *(const v4f*)(wj + t * 1024 + tid * 4);
#pragma unroll
    for (int t = 0; t < 3; ++t) {
        ws[ldsoff[t] + 0] = rg[t].x;  ws[ldsoff[t] + 2] = rg[t].y;
        ws[ldsoff[t] + 4] = rg[t].z;  ws[ldsoff[t] + 6] = rg[t].w;
    }
    __syncthreads();

    for (int kk = 0; kk < P_LEN; kk += KC) {
        const int cur   = (kk / KC) & 1;
        const bool more = (kk + KC) < P_LEN;

        // Issue next chunk's global loads early; latency hides under 48 WMMAs.
        if (more) {
#pragma unroll
            for (int t = 0; t < 3; ++t)
                rg[t] = *(const v4f*)(wj + (kk + KC) * PRE_WIN + t * 1024 + tid * 4);
            __builtin_prefetch(xr + kk + KC, 0, 0);
        }

        // Compute from current buffer. Per-lane base covers khalf row-pair select.
        const float* lb = &ws[cur * SLICE] + khalf * 96 + l15 * 2;
#pragma unroll
        for (int k = 0; k < KC; k += 4) {
            // A frag (16x4 f32): lane reads 2 consecutive floats of its row.
            v2f a = *(const v2f*)(xr + kk + k + 2 * khalf);
            // B frags: contiguous (K,K+1) pairs, one ds_load_b64 each.
            const int rbase = (k >> 1) * 96;
            v2f bb;
            bb = *(const v2f*)(lb + rbase + 0);
            c0 = __builtin_amdgcn_wmma_f32_16x16x4_f32(false, a, false, bb,
                                                       (short)0, c0, false, false);
            bb = *(const v2f*)(lb + rbase + 32);
            c1 = __builtin_amdgcn_wmma_f32_16x16x4_f32(false, a, false, bb,
                                                       (short)0, c1, false, false);
            bb = *(const v2f*)(lb + rbase + 64);
            c2 = __builtin_amdgcn_wmma_f32_16x16x4_f32(false, a, false, bb,
                                                       (short)0, c2, false, false);
        }

        // Store staged regs into the other buffer; safe: all reads of that
        // buffer completed before the barrier that opened this chunk.
        if (more) {
            float* wn = &ws[(cur ^ 1) * SLICE];
#pragma unroll
            for (int t = 0; t < 3; ++t) {
                wn[ldsoff[t] + 0] = rg[t].x;  wn[ldsoff[t] + 2] = rg[t].y;
                wn[ldsoff[t] + 4] = rg[t].z;  wn[ldsoff[t] + 6] = rg[t].w;
            }
        }
        __syncthreads();
    }

    // C/D layout: VGPR i, lanes 0-15 -> M=i, lanes 16-31 -> M=i+8; N = l15 (+16*tile).
    // out[b][w][j] at ((b*PRE_WIN + w)*Y_DIM + j).
#pragma unroll
    for (int i = 0; i < 8; ++i) {
        const int row = m0 + i + 8 * khalf;
        const size_t base = (size_t)row * PRE_WIN * Y_DIM + j;
        out[base + (size_t)( 0 + l15) * Y_DIM] = c0[i];
        out[base + (size_t)(16 + l15) * Y_DIM] = c1[i];
        out[base + (size_t)(32 + l15) * Y_DIM] = c2[i];
    }
}

extern "C" void kernel_launch(void* const* d_in, const int* in_sizes, int n_in,
                              void* d_out, int out_size, void* d_ws, size_t ws_size,
                              hipStream_t stream) {
    const float* x    = (const float*)d_in[0];  // [1024, 256, 1024]
    const float* w    = (const float*)d_in[1];  // [256, 1024, 48]
    const float* bias = (const float*)d_in[2];  // [256, 48]
    float*       out  = (float*)d_out;          // [1024, 48, 256]

    dim3 grid(B_DIM / BLK_M, Y_DIM);            // (8, 256)
    fr_bjp_jpw_bwj_wmma<<<grid, dim3(256), 0, stream>>>(x, w, bias, out);
}